// CIFLayer_86663850098892
// MI455X (gfx1250) — compile-verified
//
#include <hip/hip_runtime.h>

#define S_LEN 2048
#define BATCH 16
#define FEAT 1024
#define HID 1024
#define KW 3
#define TMAX 512
#define OUT_ELEMS (TMAX * BATCH * FEAT) /* 8388608 */
#define CH 256                          /* K-chunk width (floats) */

typedef __attribute__((ext_vector_type(2))) float v2f;
typedef __attribute__((ext_vector_type(8))) float v8f;
typedef __attribute__((ext_vector_type(4))) int   v4i;

#if defined(__has_builtin)
#if __has_builtin(__builtin_amdgcn_global_load_async_to_lds_b128) && \
    __has_builtin(__builtin_amdgcn_s_wait_asynccnt)
#define USE_ASYNC_LDS 1
#endif
#endif
#ifndef USE_ASYNC_LDS
#define USE_ASYNC_LDS 0
#endif

#define ASG __attribute__((address_space(1)))
#define ASL __attribute__((address_space(3)))

// ---------------------------------------------------------------------------
// Kernel A: conv(K=3,1024->1024) + bias + LayerNorm + ReLU + dot(lin_w) +
// sigmoid -> alpha[b,s].  One block = 32 consecutive s rows of one batch
// (M=32 so every B fragment feeds two WMMAs).  8 waves x 8 tiles cover N=1024.
// A-tile staged in LDS via async-to-LDS (ASYNCcnt) when available.
// ---------------------------------------------------------------------------
__global__ __launch_bounds__(256, 1) void alpha_kernel(
    const float* __restrict__ x, const unsigned char* __restrict__ mask,
    const float* __restrict__ conv_w, const float* __restrict__ conv_b,
    const float* __restrict__ ln_w, const float* __restrict__ ln_b,
    const float* __restrict__ lin_w, const float* __restrict__ lin_b,
    float* __restrict__ alpha)
{
    __shared__ float xs[34 * CH];           // 34 x-rows, CH features each
    __shared__ float red_sum[32], red_sq[32], red_logit[32];
    __shared__ float mu_s[32], rs_s[32];

    const int blk = blockIdx.x;
    const int b   = blk >> 6;               // batch
    const int s0  = (blk & 63) << 5;        // first of 32 s rows
    const int tid = threadIdx.x;
    const int lane = tid & 31;
    const int wv  = tid >> 5;               // wave id (N slice)
    const int hl  = lane >> 4;              // half of wave32
    const int l16 = lane & 15;

    if (tid < 32) { red_sum[tid] = 0.f; red_sq[tid] = 0.f; red_logit[tid] = 0.f; }

    v8f accL[8] = {};                       // rows m = 0..15
    v8f accH[8] = {};                       // rows m = 16..31

    for (int ic = 0; ic < FEAT / CH; ++ic) {
        __syncthreads();
        // Stage x[s0-2 .. s0+31, b, ic*CH .. +CH) -> LDS (zeros for s<0)
        for (int t = tid; t < 34 * (CH / 4); t += 256) {
            int r = t >> 6, j = t & 63;
            int s = s0 - 2 + r;
            float* ldst = xs + r * CH + j * 4;
            if (s >= 0 && s < S_LEN) {
                float* g = const_cast<float*>(x) +
                           (size_t)(s * BATCH + b) * FEAT + ic * CH + j * 4;
#if USE_ASYNC_LDS
                __builtin_amdgcn_global_load_async_to_lds_b128(
                    (ASG v4i*)g, (ASL v4i*)ldst, 0, 0);
#else
                *(float4*)ldst = *(const float4*)g;
#endif
            } else {
                *(float4*)ldst = make_float4(0.f, 0.f, 0.f, 0.f);
            }
        }
#if USE_ASYNC_LDS
        __builtin_amdgcn_s_wait_asynccnt(0);
#endif
        __syncthreads();
        for (int k = 0; k < KW; ++k) {
            // A[m, c] for this (k, ic) chunk lives at xs[(m+k)*CH + i']
            const float* AbL = xs + (l16 + k) * CH + hl * 2;
            const float* AbH = AbL + 16 * CH;
            const float* B0 = conv_w + ((size_t)(k * FEAT + ic * CH) + hl * 2) * HID
                              + wv * 128 + l16;
            #pragma unroll 1
            for (int ip = 0; ip < CH; ip += 4) {
                v2f AL, AH;
                AL.x = AbL[ip];
                AL.y = AbL[ip + 1];
                AH.x = AbH[ip];
                AH.y = AbH[ip + 1];
                const float* bp = B0 + (size_t)ip * HID;
                #pragma unroll
                for (int t = 0; t < 8; ++t) {
                    v2f Bf;
                    Bf.x = bp[t * 16];
                    Bf.y = bp[t * 16 + HID];
                    accL[t] = __builtin_amdgcn_wmma_f32_16x16x4_f32(
                        false, AL, false, Bf, (short)0, accL[t], false, false);
                    accH[t] = __builtin_amdgcn_wmma_f32_16x16x4_f32(
                        false, AH, false, Bf, (short)0, accH[t], false, false);
                }
            }
        }
    }

    // accL[t][r]: row m = hl*8 + r;  accH[t][r]: row m = 16 + hl*8 + r
    // col n = wv*128 + t*16 + l16
    float ps[8], pq[8];

    // ---- bias + LN stats, rows 0..15 ----
    #pragma unroll
    for (int r = 0; r < 8; ++r) { ps[r] = 0.f; pq[r] = 0.f; }
    #pragma unroll
    for (int t = 0; t < 8; ++t) {
        float cb = conv_b[wv * 128 + t * 16 + l16];
        #pragma unroll
        for (int r = 0; r < 8; ++r) {
            float v = accL[t][r] + cb;
            accL[t][r] = v;
            ps[r] += v; pq[r] += v * v;
        }
    }
    #pragma unroll
    for (int r = 0; r < 8; ++r) {
        #pragma unroll
        for (int m = 8; m >= 1; m >>= 1) {
            ps[r] += __shfl_xor(ps[r], m, 32);
            pq[r] += __shfl_xor(pq[r], m, 32);
        }
    }
    if (l16 == 0) {
        #pragma unroll
        for (int r = 0; r < 8; ++r) {
            atomicAdd(&red_sum[hl * 8 + r], ps[r]);
            atomicAdd(&red_sq[hl * 8 + r],  pq[r]);
        }
    }
    // ---- bias + LN stats, rows 16..31 ----
    #pragma unroll
    for (int r = 0; r < 8; ++r) { ps[r] = 0.f; pq[r] = 0.f; }
    #pragma unroll
    for (int t = 0; t < 8; ++t) {
        float cb = conv_b[wv * 128 + t * 16 + l16];
        #pragma unroll
        for (int r = 0; r < 8; ++r) {
            float v = accH[t][r] + cb;
            accH[t][r] = v;
            ps[r] += v; pq[r] += v * v;
        }
    }
    #pragma unroll
    for (int r = 0; r < 8; ++r) {
        #pragma unroll
        for (int m = 8; m >= 1; m >>= 1) {
            ps[r] += __shfl_xor(ps[r], m, 32);
            pq[r] += __shfl_xor(pq[r], m, 32);
        }
    }
    if (l16 == 0) {
        #pragma unroll
        for (int r = 0; r < 8; ++r) {
            atomicAdd(&red_sum[16 + hl * 8 + r], ps[r]);
            atomicAdd(&red_sq[16 + hl * 8 + r],  pq[r]);
        }
    }
    __syncthreads();
    if (tid < 32) {
        float mu  = red_sum[tid] * (1.f / HID);
        float var = red_sq[tid] * (1.f / HID) - mu * mu;
        mu_s[tid] = mu;
        rs_s[tid] = rsqrtf(var + 1e-5f);
    }
    __syncthreads();

    float pl[8];
    // ---- logit, rows 0..15 ----
    #pragma unroll
    for (int r = 0; r < 8; ++r) pl[r] = 0.f;
    #pragma unroll
    for (int t = 0; t < 8; ++t) {
        int n = wv * 128 + t * 16 + l16;
        float lnw = ln_w[n], lnb = ln_b[n], lw = lin_w[n];
        #pragma unroll
        for (int r = 0; r < 8; ++r) {
            int row = hl * 8 + r;
            float hn = (accL[t][r] - mu_s[row]) * rs_s[row] * lnw + lnb;
            hn = fmaxf(hn, 0.f);
            pl[r] += hn * lw;
        }
    }
    #pragma unroll
    for (int r = 0; r < 8; ++r) {
        #pragma unroll
        for (int m = 8; m >= 1; m >>= 1) pl[r] += __shfl_xor(pl[r], m, 32);
    }
    if (l16 == 0) {
        #pragma unroll
        for (int r = 0; r < 8; ++r) atomicAdd(&red_logit[hl * 8 + r], pl[r]);
    }
    // ---- logit, rows 16..31 ----
    #pragma unroll
    for (int r = 0; r < 8; ++r) pl[r] = 0.f;
    #pragma unroll
    for (int t = 0; t < 8; ++t) {
        int n = wv * 128 + t * 16 + l16;
        float lnw = ln_w[n], lnb = ln_b[n], lw = lin_w[n];
        #pragma unroll
        for (int r = 0; r < 8; ++r) {
            int row = 16 + hl * 8 + r;
            float hn = (accH[t][r] - mu_s[row]) * rs_s[row] * lnw + lnb;
            hn = fmaxf(hn, 0.f);
            pl[r] += hn * lw;
        }
    }
    #pragma unroll
    for (int r = 0; r < 8; ++r) {
        #pragma unroll
        for (int m = 8; m >= 1; m >>= 1) pl[r] += __shfl_xor(pl[r], m, 32);
    }
    if (l16 == 0) {
        #pragma unroll
        for (int r = 0; r < 8; ++r) atomicAdd(&red_logit[16 + hl * 8 + r], pl[r]);
    }
    __syncthreads();
    if (tid < 32) {
        int s = s0 + tid;
        float logit = red_logit[tid] + lin_b[0];
        float al = 1.f / (1.f + expf(-logit));
        if (mask[b * S_LEN + s]) al = 0.f;
        alpha[b * S_LEN + s] = al;
    }
}

// ---------------------------------------------------------------------------
__global__ void zero_kernel(float4* __restrict__ out4)
{
    size_t i = (size_t)blockIdx.x * blockDim.x + threadIdx.x;
    out4[i] = make_float4(0.f, 0.f, 0.f, 0.f);
}

// per-batch alpha sum -> alpha_sum output + scale[b]; also emit lengths
__global__ __launch_bounds__(256) void sum_kernel(
    const float* __restrict__ alpha, const int* __restrict__ tlen,
    float* __restrict__ out, float* __restrict__ scale)
{
    int b = blockIdx.x, tid = threadIdx.x;
    float p = 0.f;
    for (int i = tid; i < S_LEN; i += 256) p += alpha[b * S_LEN + i];
    #pragma unroll
    for (int m = 16; m >= 1; m >>= 1) p += __shfl_xor(p, m, 32);
    __shared__ float red[8];
    if ((tid & 31) == 0) red[tid >> 5] = p;
    __syncthreads();
    if (tid == 0) {
        float asum = 0.f;
        for (int w = 0; w < 8; ++w) asum += red[w];
        float desired = (float)tlen[b] + 1e-4f;   // BETA=1, EPS=1e-4
        scale[b] = desired / asum;
        out[OUT_ELEMS + BATCH + b] = asum;
        ((int*)out)[OUT_ELEMS + b] = tlen[b];
    }
}

// per-batch inclusive scan of a = alpha*scale
__global__ __launch_bounds__(256) void scan_kernel(
    const float* __restrict__ alpha, const float* __restrict__ scale,
    float* __restrict__ csum)
{
    int b = blockIdx.x, tid = threadIdx.x;
    int lane = tid & 31, wv = tid >> 5;
    __shared__ float wsum[8];
    __shared__ float carry;
    if (tid == 0) carry = 0.f;
    float sc = scale[b];
    for (int seg = 0; seg < S_LEN; seg += 256) {
        __syncthreads();
        float v = alpha[b * S_LEN + seg + tid] * sc;
        #pragma unroll
        for (int d = 1; d < 32; d <<= 1) {
            float n = __shfl_up(v, d, 32);
            if (lane >= d) v += n;
        }
        if (lane == 31) wsum[wv] = v;
        __syncthreads();
        if (tid == 0) {
            float run = 0.f;
            for (int w = 0; w < 8; ++w) { float t = wsum[w]; wsum[w] = run; run += t; }
        }
        __syncthreads();
        v += wsum[wv] + carry;
        csum[b * S_LEN + seg + tid] = v;
        __syncthreads();
        if (tid == 255) carry = v;
    }
}

// CIF scatter: one block per (b,s); up to 4 weighted atomic row-adds into
// the transposed (T,B,C) output.  Row index TMAX (=T) is the dropped row.
__global__ __launch_bounds__(256) void scatter_kernel(
    const float* __restrict__ x, const float* __restrict__ alpha,
    const float* __restrict__ scale, const float* __restrict__ csum,
    float* __restrict__ out)
{
    int bs = blockIdx.x;
    int b = bs >> 11, s = bs & (S_LEN - 1);
    int tid = threadIdx.x;
    __shared__ float sw[4];
    __shared__ int   si[4];
    if (tid == 0) {
        float c1 = csum[b * S_LEN + s];
        float c0 = (s > 0) ? csum[b * S_LEN + s - 1] : 0.f;
        int ri = (int)floorf(c1); ri = ri < 0 ? 0 : (ri > TMAX ? TMAX : ri);
        int li = 0;
        if (s > 0) { li = (int)floorf(c0); li = li < 0 ? 0 : (li > TMAX ? TMAX : li); }
        int fire  = ri - li;
        int extra = (fire - 1 > 0) ? fire - 1 : 0;
        float a  = alpha[b * S_LEN + s] * scale[b];
        float rw = (fire > 0) ? (c1 - (float)ri) : 0.f;
        float lw = a - rw - (float)extra;
        si[0] = ri; sw[0] = rw;
        si[1] = li; sw[1] = lw;
        int t1 = (li + 1 < TMAX) ? li + 1 : TMAX;
        int t2 = (li + 2 < TMAX) ? li + 2 : TMAX;
        si[2] = t1; sw[2] = (extra > 0) ? 1.f : 0.f;
        si[3] = t2; sw[3] = (extra > 1) ? 1.f : 0.f;
    }
    __syncthreads();
    float4 xv = *(const float4*)(x + (size_t)(s * BATCH + b) * FEAT + tid * 4);
    #pragma unroll
    for (int j = 0; j < 4; ++j) {
        float w = sw[j]; int tt = si[j];
        if (w != 0.f && tt < TMAX) {
            float* o = out + (size_t)(tt * BATCH + b) * FEAT + tid * 4;
            atomicAdd(o + 0, w * xv.x);
            atomicAdd(o + 1, w * xv.y);
            atomicAdd(o + 2, w * xv.z);
            atomicAdd(o + 3, w * xv.w);
        }
    }
}

// ---------------------------------------------------------------------------
extern "C" void kernel_launch(void* const* d_in, const int* in_sizes, int n_in,
                              void* d_out, int out_size, void* d_ws, size_t ws_size,
                              hipStream_t stream)
{
    (void)in_sizes; (void)n_in; (void)out_size; (void)ws_size;
    const float*         x      = (const float*)d_in[0];
    const unsigned char* mask   = (const unsigned char*)d_in[1];
    const int*           tlen   = (const int*)d_in[2];
    const float*         conv_w = (const float*)d_in[3];
    const float*         conv_b = (const float*)d_in[4];
    const float*         ln_w   = (const float*)d_in[5];
    const float*         ln_b   = (const float*)d_in[6];
    const float*         lin_w  = (const float*)d_in[7];
    const float*         lin_b  = (const float*)d_in[8];
    float* out   = (float*)d_out;
    float* alpha = (float*)d_ws;
    float* csum  = alpha + BATCH * S_LEN;
    float* scale = csum + BATCH * S_LEN;

    alpha_kernel<<<BATCH * (S_LEN / 32), 256, 0, stream>>>(
        x, mask, conv_w, conv_b, ln_w, ln_b, lin_w, lin_b, alpha);
    zero_kernel<<<OUT_ELEMS / 4 / 256, 256, 0, stream>>>((float4*)d_out);
    sum_kernel<<<BATCH, 256, 0, stream>>>(alpha, tlen, out, scale);
    scan_kernel<<<BATCH, 256, 0, stream>>>(alpha, scale, csum);
    scatter_kernel<<<BATCH * S_LEN, 256, 0, stream>>>(x, alpha, scale, csum, out);
}